// Neural_44770739094212
// MI455X (gfx1250) — compile-verified
//
#include <hip/hip_runtime.h>

// ---------------------------------------------------------------------------
// Problem constants (match reference)
// ---------------------------------------------------------------------------
#define NC 4          // channels
#define NBATCH 32     // images per channel
#define IH 512
#define IW 512
#define TS 32         // output tile (TS x TS)
#define LT 34         // tile + 1-halo on each side
#define LTC (LT * LT) // 1156 cells per LDS plane
#define NTHREADS 256

// ---------------------------------------------------------------------------
// CDNA5 async global->LDS staging (gfx1250), ASYNCcnt-tracked.
// Builtin signature (probe-verified): (AS1 int*, AS3 int*, imm off, imm cpol)
// ---------------------------------------------------------------------------
#if defined(__has_builtin)
#if __has_builtin(__builtin_amdgcn_global_load_async_to_lds_b32)
#define HAVE_ASYNC_LDS 1
#endif
#endif

typedef __attribute__((address_space(1))) int as1_int;
typedef __attribute__((address_space(3))) int as3_int;

__device__ __forceinline__ void stage_b32(const float* g, float* l) {
#ifdef HAVE_ASYNC_LDS
  __builtin_amdgcn_global_load_async_to_lds_b32(
      (as1_int*)g, (as3_int*)l, /*imm offset*/ 0, /*cpol*/ 0);
#else
  *l = *g;
#endif
}

__device__ __forceinline__ void wait_async_lds() {
#ifdef HAVE_ASYNC_LDS
#if __has_builtin(__builtin_amdgcn_s_wait_asynccnt)
  __builtin_amdgcn_s_wait_asynccnt(0);
#else
  asm volatile("s_wait_asynccnt 0" ::: "memory");
#endif
#endif
}

// Hardware transcendentals (v_log_f32 / v_exp_f32), ~1 ulp.
__device__ __forceinline__ float flog2(float x) {
#if defined(__has_builtin) && __has_builtin(__builtin_amdgcn_logf)
  return __builtin_amdgcn_logf(x);
#else
  return __log2f(x);
#endif
}
__device__ __forceinline__ float fexp2(float x) {
#if defined(__has_builtin) && __has_builtin(__builtin_amdgcn_exp2f)
  return __builtin_amdgcn_exp2f(x);
#else
  return exp2f(x);
#endif
}

// Streaming (non-temporal) store: output is write-once, keep it out of L2 so
// the re-read source halos stay resident (src+out = 256MB > 192MB L2).
__device__ __forceinline__ void store_nt(float* p, float v) {
#if defined(__has_builtin) && __has_builtin(__builtin_nontemporal_store)
  __builtin_nontemporal_store(v, p);
#else
  *p = v;
#endif
}

// ---------------------------------------------------------------------------
// 3x3 correlation over a 4-row column strip with a rolling 3-row window.
// Scalar FMAs: chosen over pk_fma (VOP3P pair-marshalling moves cost
// more than the packed win); gfx1250 VOPD dual-issue pairs these for free.
// out(oy,ox) += sum_{dy,dx} P[oy+dy][ox+dx] * k[dy][dx]
// ---------------------------------------------------------------------------
__device__ __forceinline__ void conv_acc(const float* __restrict__ P, int ox,
                                         int oyb, const float* __restrict__ kg,
                                         float acc[4]) {
  float kk[9];
#pragma unroll
  for (int t = 0; t < 9; ++t) kk[t] = kg[t];  // uniform -> scalar loads

  const float* row = P + oyb * LT + ox;
  float w0[3], w1[3], w2[3];
#pragma unroll
  for (int c = 0; c < 3; ++c) {
    w0[c] = row[c];
    w1[c] = row[LT + c];
  }
#pragma unroll
  for (int r = 0; r < 4; ++r) {
#pragma unroll
    for (int c = 0; c < 3; ++c) w2[c] = row[(2 + r) * LT + c];
    acc[r] += w0[0] * kk[0] + w0[1] * kk[1] + w0[2] * kk[2] +
              w1[0] * kk[3] + w1[1] * kk[4] + w1[2] * kk[5] +
              w2[0] * kk[6] + w2[1] * kk[7] + w2[2] * kk[8];
#pragma unroll
    for (int c = 0; c < 3; ++c) {
      w0[c] = w1[c];
      w1[c] = w2[c];
    }
  }
}

// ---------------------------------------------------------------------------
// Fused bleed-correction kernel: one pass over the data.
// sources: [4][32][512][512], kernels: [12][3][3], out: [4][32][512][512]
// ---------------------------------------------------------------------------
__global__ __launch_bounds__(NTHREADS) void bleed_correct_kernel(
    const float* __restrict__ src, const float* __restrict__ ker,
    float* __restrict__ out) {
  __shared__ float S[NC][LTC];  // staged source tiles (with halo)
  __shared__ float T[6][LTC];   // inter-term planes (with halo)

  const int tid = threadIdx.x;
  const int x0 = blockIdx.x * TS;
  const int y0 = blockIdx.y * TS;
  const int n = blockIdx.z;

  // ---- Phase 1: stage 4 haloed source tiles into LDS (async DMA) ----------
  for (int k = tid; k < NC * LTC; k += NTHREADS) {
    const int p = k / LTC;
    const int rem = k - p * LTC;
    const int r = rem / LT;
    const int c = rem - r * LT;
    const int gy = y0 - 1 + r;
    const int gx = x0 - 1 + c;
    float* dst = &S[p][rem];
    if ((unsigned)gy < (unsigned)IH && (unsigned)gx < (unsigned)IW) {
      const float* g =
          src + ((((size_t)p * NBATCH + n) * IH + gy) * IW + gx);
      stage_b32(g, dst);  // ASYNCcnt-tracked global->LDS
    } else {
      *dst = 0.0f;  // SAME padding = zeros (disjoint cells, no ordering hazard)
    }
  }
  wait_async_lds();
  __syncthreads();

  // ---- Phase 2: inter-term planes: (nb^0.5 * src)^(2/3)
  //      = exp2(log2(nb)/3 + 2*log2(src)/3); share the 4 log2 per cell ------
  {
    const float c13 = 1.0f / 3.0f;
    const float c23 = 2.0f / 3.0f;
    for (int k = tid; k < LTC; k += NTHREADS) {
      const float L0 = flog2(S[0][k]);
      const float L1 = flog2(S[1][k]);
      const float L2 = flog2(S[2][k]);
      const float L3 = flog2(S[3][k]);
      T[0][k] = fexp2(c13 * L1 + c23 * L0);  // nb=s1, src=s0
      T[1][k] = fexp2(c13 * L0 + c23 * L1);  // nb=s0, src=s1
      T[2][k] = fexp2(c13 * L2 + c23 * L1);  // nb=s2, src=s1
      T[3][k] = fexp2(c13 * L1 + c23 * L2);  // nb=s1, src=s2
      T[4][k] = fexp2(c13 * L3 + c23 * L2);  // nb=s3, src=s2
      T[5][k] = fexp2(c13 * L2 + c23 * L3);  // nb=s2, src=s3
    }
  }
  __syncthreads();

  // ---- Phase 3: 12 convs, subtract bleed, streaming stores ---------------
  const int ox = tid & (TS - 1);   // 0..31 (coalesced x)
  const int oyb = (tid >> 5) * 4;  // 4-row strip per thread

  float acc[4];

#define EMIT_CHANNEL(CH, BODY)                                                \
  {                                                                           \
    _Pragma("unroll") for (int r = 0; r < 4; ++r) acc[r] = 0.0f;              \
    BODY;                                                                     \
    _Pragma("unroll") for (int r = 0; r < 4; ++r) {                           \
      const int oy = oyb + r;                                                 \
      const float v = S[CH][(oy + 1) * LT + (ox + 1)] - acc[r];               \
      store_nt(                                                               \
          &out[(((size_t)(CH)*NBATCH + n) * IH + (y0 + oy)) * IW + (x0 + ox)],\
          v);                                                                 \
    }                                                                         \
  }

  // channel 0: bleed = conv(s1,k0) + conv(t(1->0),k1)
  EMIT_CHANNEL(0, {
    conv_acc(S[1], ox, oyb, ker + 0 * 9, acc);
    conv_acc(T[0], ox, oyb, ker + 1 * 9, acc);
  });
  // channel 1: conv(s0,k2)+conv(t(0->1),k3)+conv(s2,k4)+conv(t(2->1),k5)
  EMIT_CHANNEL(1, {
    conv_acc(S[0], ox, oyb, ker + 2 * 9, acc);
    conv_acc(T[1], ox, oyb, ker + 3 * 9, acc);
    conv_acc(S[2], ox, oyb, ker + 4 * 9, acc);
    conv_acc(T[2], ox, oyb, ker + 5 * 9, acc);
  });
  // channel 2: conv(s1,k6)+conv(t(1->2),k7)+conv(s3,k8)+conv(t(3->2),k9)
  EMIT_CHANNEL(2, {
    conv_acc(S[1], ox, oyb, ker + 6 * 9, acc);
    conv_acc(T[3], ox, oyb, ker + 7 * 9, acc);
    conv_acc(S[3], ox, oyb, ker + 8 * 9, acc);
    conv_acc(T[4], ox, oyb, ker + 9 * 9, acc);
  });
  // channel 3: conv(s2,k10)+conv(t(2->3),k11)
  EMIT_CHANNEL(3, {
    conv_acc(S[2], ox, oyb, ker + 10 * 9, acc);
    conv_acc(T[5], ox, oyb, ker + 11 * 9, acc);
  });
#undef EMIT_CHANNEL
}

// ---------------------------------------------------------------------------
// Host launch
// ---------------------------------------------------------------------------
extern "C" void kernel_launch(void* const* d_in, const int* in_sizes, int n_in,
                              void* d_out, int out_size, void* d_ws,
                              size_t ws_size, hipStream_t stream) {
  (void)in_sizes;
  (void)n_in;
  (void)d_ws;
  (void)ws_size;
  (void)out_size;
  const float* sources = (const float*)d_in[0];  // [4][32][512][512] f32
  const float* kernels = (const float*)d_in[1];  // [12][3][3] f32
  float* out = (float*)d_out;                    // [4][32][512][512] f32

  dim3 grid(IW / TS, IH / TS, NBATCH);  // 16 x 16 x 32 = 8192 workgroups
  bleed_correct_kernel<<<grid, NTHREADS, 0, stream>>>(sources, kernels, out);
}